// GAT_12223476924905
// MI455X (gfx1250) — compile-verified
//
#include <hip/hip_runtime.h>
#include <hip/hip_bf16.h>
#include <math.h>

#define NN 100000
#define NEG_SLOPE 0.2f

typedef float v2f __attribute__((ext_vector_type(2)));
typedef float v8f __attribute__((ext_vector_type(8)));

// ---------------------------------------------------------------------------
// FP32 WMMA GEMM: D[M,Nc] = A[M,K] * Bt[Nc,K]^T, with B pre-transposed so both
// operand fetches are contiguous 8-byte loads.  One wave computes TWO adjacent
// 16x16 output tiles along N, reusing the A fragment across two WMMAs.
// A 16x4 layout (ISA 7.12.2): lanes 0-15 hold M=0..15; VGPR0={K0|K2},
// VGPR1={K1|K3}.  B mirrored (lane = N, lo/hi half = K pair).
// C/D: VGPR i -> row i (lanes 0-15) / row 8+i (lanes 16-31), col = lane&15.
// ---------------------------------------------------------------------------
template <int K>
__global__ void gemm_f32_wmma(const float* __restrict__ A, const float* __restrict__ Bt,
                              float* __restrict__ D, int M, int Nc) {
  int wave   = blockIdx.x * (blockDim.x >> 5) + (threadIdx.x >> 5);
  int lane   = threadIdx.x & 31;
  int pairsN = Nc >> 5;                    // two 16-wide tiles per wave
  int tileM  = wave / pairsN;
  int pN     = wave - tileM * pairsN;
  if (tileM * 16 >= M) return;             // wave-uniform exit: EXEC stays full
  const int row0 = tileM << 4, col0 = pN << 5;
  const int l15 = lane & 15;
  const int hi  = lane >> 4;               // 0: K pair {0,1}, 1: K pair {2,3}
  const float* __restrict__ Ar = A  + (size_t)(row0 + l15) * K + 2 * hi;
  const float* __restrict__ B0 = Bt + (size_t)(col0 + l15) * K + 2 * hi;
  const float* __restrict__ B1 = B0 + (size_t)16 * K;
  v8f acc0 = {}, acc1 = {};
#pragma unroll 4
  for (int k = 0; k < K; k += 4) {
    v2f a  = *reinterpret_cast<const v2f*>(Ar + k);
    v2f b0 = *reinterpret_cast<const v2f*>(B0 + k);
    v2f b1 = *reinterpret_cast<const v2f*>(B1 + k);
    acc0 = __builtin_amdgcn_wmma_f32_16x16x4_f32(false, a, false, b0, (short)0, acc0,
                                                 false, false);
    acc1 = __builtin_amdgcn_wmma_f32_16x16x4_f32(false, a, false, b1, (short)0, acc1,
                                                 false, false);
  }
  float* __restrict__ Dp = D + (size_t)(row0 + hi * 8) * Nc + col0 + l15;
#pragma unroll
  for (int i = 0; i < 8; ++i) {
    Dp[(size_t)i * Nc]      = acc0[i];
    Dp[(size_t)i * Nc + 16] = acc1[i];
  }
}

// one-off weight transpose: Bt[n*K + k] = B[k*Nc + n]
__global__ void transpose_w(const float* __restrict__ B, float* __restrict__ Bt,
                            int K, int Nc) {
  int idx = blockIdx.x * blockDim.x + threadIdx.x;
  if (idx >= K * Nc) return;
  int k = idx / Nc, n = idx - k * Nc;
  Bt[n * K + k] = B[idx];
}

// ---------------------------------------------------------------------------
// Edge list with self loops (int32)
// ---------------------------------------------------------------------------
__global__ void build_edges(const int* __restrict__ ei, int* __restrict__ src,
                            int* __restrict__ dst, int Ein, int Etot) {
  int i = blockIdx.x * blockDim.x + threadIdx.x;
  if (i >= Etot) return;
  if (i < Ein) { src[i] = ei[i]; dst[i] = ei[Ein + i]; }
  else         { src[i] = i - Ein; dst[i] = i - Ein; }
}

// per-(node,head) attention dot products: as/ad[n*H+h] = <h_row, att_{s,d}[h]>
template <int H, int C>
__global__ void att_scores(const float* __restrict__ h, const float* __restrict__ ws,
                           const float* __restrict__ wd, float* __restrict__ as,
                           float* __restrict__ ad, int total) {
  int idx = blockIdx.x * blockDim.x + threadIdx.x;
  if (idx >= total) return;
  int n = idx / H, hh = idx - n * H;
  const float* __restrict__ row = h + ((size_t)n * H + hh) * C;
  float s = 0.f, d = 0.f;
#pragma unroll
  for (int c = 0; c < C; ++c) {
    float v = row[c];
    s += v * ws[hh * C + c];
    d += v * wd[hh * C + c];
  }
  as[idx] = s;
  ad[idx] = d;
}

// order-preserving float<->uint transform for atomic max over signed floats
__device__ __forceinline__ unsigned f2ord(float f) {
  unsigned u = __float_as_uint(f);
  return (u & 0x80000000u) ? ~u : (u | 0x80000000u);
}
__device__ __forceinline__ float ord2f(unsigned u) {
  return (u & 0x80000000u) ? __uint_as_float(u ^ 0x80000000u)
                           : __uint_as_float(~u);
}

// pass A: e = leaky(a_src[src]+a_dst[dst]); store; segment max via atomic
template <int H>
__global__ void edge_scores(const int* __restrict__ src, const int* __restrict__ dst,
                            const float* __restrict__ as, const float* __restrict__ ad,
                            float* __restrict__ ebuf, unsigned* __restrict__ mord,
                            int total) {
  int idx = blockIdx.x * blockDim.x + threadIdx.x;
  if (idx >= total) return;
  int e = idx / H, hh = idx - e * H;      // H is a power of two -> shifts
  float v = as[src[e] * H + hh] + ad[dst[e] * H + hh];
  v = v > 0.f ? v : NEG_SLOPE * v;
  ebuf[idx] = v;
  atomicMax(mord + dst[e] * H + hh, f2ord(v));
}

// pass B: ex = exp(e - m[dst]); store; segment sum via atomic
template <int H>
__global__ void edge_exp(const int* __restrict__ dst, float* __restrict__ ebuf,
                         const unsigned* __restrict__ mord, float* __restrict__ den,
                         int total) {
  int idx = blockIdx.x * blockDim.x + threadIdx.x;
  if (idx >= total) return;
  int e = idx / H, hh = idx - e * H;
  int dh = dst[e] * H + hh;
  float ex = __expf(ebuf[idx] - ord2f(mord[dh]));
  ebuf[idx] = ex;
  atomicAdd(den + dh, ex);
}

__global__ void recip_den(float* __restrict__ den, int total) {
  int idx = blockIdx.x * blockDim.x + threadIdx.x;
  if (idx >= total) return;
  den[idx] = 1.0f / (den[idx] + 1e-16f);
}

// pass C: out[dst] += alpha * h[src], one thread per (edge, float4 chunk)
template <int H, int C, int HC>
__global__ void edge_agg(const int* __restrict__ src, const int* __restrict__ dst,
                         const float* __restrict__ h, const float* __restrict__ ex,
                         const float* __restrict__ rden, float* __restrict__ out,
                         int total) {
  int idx = blockIdx.x * blockDim.x + threadIdx.x;
  if (idx >= total) return;
  constexpr int chunks = HC >> 2;
  int e  = idx / chunks;                   // power of two -> shift
  int cf = idx - e * chunks;
  int s = src[e], d = dst[e];
  int hh = (cf << 2) / C;                  // compile-time divisor
  float alpha = ex[e * H + hh] * rden[d * H + hh];
  float4 v = reinterpret_cast<const float4*>(h + (size_t)s * HC)[cf];
  float* op = out + (size_t)d * HC + (cf << 2);
  atomicAdd(op + 0, alpha * v.x);
  atomicAdd(op + 1, alpha * v.y);
  atomicAdd(op + 2, alpha * v.z);
  atomicAdd(op + 3, alpha * v.w);
}

// bias (+ optional ELU), in place
template <int F, bool ELU>
__global__ void bias_act(float* __restrict__ o, const float* __restrict__ b, int total) {
  int idx = blockIdx.x * blockDim.x + threadIdx.x;
  if (idx >= total) return;
  float v = o[idx] + b[idx & (F - 1)];     // F is a power of two
  if (ELU) v = v > 0.f ? v : (__expf(v) - 1.0f);
  o[idx] = v;
}

// ---------------------------------------------------------------------------
extern "C" void kernel_launch(void* const* d_in, const int* in_sizes, int n_in,
                              void* d_out, int out_size, void* d_ws, size_t ws_size,
                              hipStream_t stream) {
  (void)n_in; (void)out_size; (void)ws_size;
  const float* x    = (const float*)d_in[0];
  const int*   ei   = (const int*)  d_in[1];
  const float* W1   = (const float*)d_in[2];
  const float* as1w = (const float*)d_in[3];
  const float* ad1w = (const float*)d_in[4];
  const float* b1   = (const float*)d_in[5];
  const float* W2   = (const float*)d_in[6];
  const float* as2w = (const float*)d_in[7];
  const float* ad2w = (const float*)d_in[8];
  const float* b2   = (const float*)d_in[9];
  float* out = (float*)d_out;

  const int Ein  = in_sizes[1] / 2;
  const int Etot = Ein + NN;

  char* wsb = (char*)d_ws;
  size_t off = 0;
  auto take = [&](size_t bytes) -> void* {
    void* p = wsb + off;
    off = (off + bytes + 255) & ~(size_t)255;
    return p;
  };
  int*      src  = (int*)     take((size_t)Etot * 4);
  int*      dst  = (int*)     take((size_t)Etot * 4);
  float*    h1   = (float*)   take((size_t)NN * 128 * 4);  // layer-2 reuses as h2
  float*    o1   = (float*)   take((size_t)NN * 128 * 4);
  float*    asrc = (float*)   take((size_t)NN * 8 * 4);
  float*    adst = (float*)   take((size_t)NN * 8 * 4);
  unsigned* mord = (unsigned*)take((size_t)NN * 8 * 4);
  float*    den  = (float*)   take((size_t)NN * 8 * 4);
  float*    ebuf = (float*)   take((size_t)Etot * 8 * 4);
  float*    Wt1  = (float*)   take((size_t)128 * 128 * 4);
  float*    Wt2  = (float*)   take((size_t)128 * 64 * 4);
  float*    h2   = h1;

  const int T = 256;
  build_edges<<<(Etot + T - 1) / T, T, 0, stream>>>(ei, src, dst, Ein, Etot);
  transpose_w<<<(128 * 128 + T - 1) / T, T, 0, stream>>>(W1, Wt1, 128, 128);
  transpose_w<<<(128 * 64 + T - 1) / T, T, 0, stream>>>(W2, Wt2, 128, 64);

  // ---------------- Layer 1: Fin=128, H=8, C=16 ----------------
  {
    int waves = (NN / 16) * (128 / 32);    // two N-tiles per wave
    gemm_f32_wmma<128><<<(waves + 3) / 4, 128, 0, stream>>>(x, Wt1, h1, NN, 128);
  }
  att_scores<8, 16><<<(NN * 8 + T - 1) / T, T, 0, stream>>>(h1, as1w, ad1w, asrc, adst, NN * 8);
  (void)hipMemsetAsync(mord, 0, (size_t)NN * 8 * 4, stream);
  (void)hipMemsetAsync(den,  0, (size_t)NN * 8 * 4, stream);
  (void)hipMemsetAsync(o1,   0, (size_t)NN * 128 * 4, stream);
  edge_scores<8><<<(Etot * 8 + T - 1) / T, T, 0, stream>>>(src, dst, asrc, adst, ebuf, mord, Etot * 8);
  edge_exp<8><<<(Etot * 8 + T - 1) / T, T, 0, stream>>>(dst, ebuf, mord, den, Etot * 8);
  recip_den<<<(NN * 8 + T - 1) / T, T, 0, stream>>>(den, NN * 8);
  edge_agg<8, 16, 128><<<(Etot * 32 + T - 1) / T, T, 0, stream>>>(src, dst, h1, ebuf, den, o1, Etot * 32);
  bias_act<128, true><<<(NN * 128 + T - 1) / T, T, 0, stream>>>(o1, b1, NN * 128);

  // ---------------- Layer 2: Fin=128, H=1, C=64 ----------------
  {
    int waves = (NN / 16) * (64 / 32);
    gemm_f32_wmma<128><<<(waves + 3) / 4, 128, 0, stream>>>(o1, Wt2, h2, NN, 64);
  }
  att_scores<1, 64><<<(NN + T - 1) / T, T, 0, stream>>>(h2, as2w, ad2w, asrc, adst, NN);
  (void)hipMemsetAsync(mord, 0, (size_t)NN * 4, stream);
  (void)hipMemsetAsync(den,  0, (size_t)NN * 4, stream);
  (void)hipMemsetAsync(out,  0, (size_t)NN * 64 * 4, stream);
  edge_scores<1><<<(Etot + T - 1) / T, T, 0, stream>>>(src, dst, asrc, adst, ebuf, mord, Etot);
  edge_exp<1><<<(Etot + T - 1) / T, T, 0, stream>>>(dst, ebuf, mord, den, Etot);
  recip_den<<<(NN + T - 1) / T, T, 0, stream>>>(den, NN);
  edge_agg<1, 64, 64><<<(Etot * 16 + T - 1) / T, T, 0, stream>>>(src, dst, h2, ebuf, den, out, Etot * 16);
  bias_act<64, false><<<(NN * 64 + T - 1) / T, T, 0, stream>>>(out, b2, NN * 64);
}